// GNNPrefetch_11398843204124
// MI455X (gfx1250) — compile-verified
//
#include <hip/hip_runtime.h>

typedef float v2f __attribute__((ext_vector_type(2)));
typedef float v8f __attribute__((ext_vector_type(8)));

// ---------------------------------------------------------------------------
// Scatter-add of neighbor rows: one wave per edge (grid-stride over edges).
// Each lane handles 4 consecutive floats (float4 read, 4x f32 atomics).
// Lane 0 bumps the degree counter. Prefetch the next edge's source row.
// ---------------------------------------------------------------------------
__global__ void __launch_bounds__(256)
sage_scatter(const float* __restrict__ x, const int* __restrict__ src,
             const int* __restrict__ dst, float* __restrict__ agg,
             float* __restrict__ cnt, int E, int d, int do_count)
{
    const int lane = threadIdx.x & 31;
    const int wid  = blockIdx.x * (blockDim.x >> 5) + (threadIdx.x >> 5);
    const int nw   = gridDim.x * (blockDim.x >> 5);
    const int d4   = d >> 2;

    for (int e = wid; e < E; e += nw) {
        const int s = src[e];
        const int t = dst[e];

        const int e2 = e + nw;
        if (e2 < E) {
            // emits global_prefetch_b8: warm next source row into cache
            __builtin_prefetch(&x[(size_t)src[e2] * d + lane * 4], 0, 1);
        }

        const float4* xs = (const float4*)(x + (size_t)s * d);
        float*        at = agg + (size_t)t * d;

        for (int c = lane; c < d4; c += 32) {
            float4 v = xs[c];
            atomicAdd(at + 4 * c + 0, v.x);
            atomicAdd(at + 4 * c + 1, v.y);
            atomicAdd(at + 4 * c + 2, v.z);
            atomicAdd(at + 4 * c + 3, v.w);
        }
        if (do_count && lane == 0) atomicAdd(cnt + t, 1.0f);
    }
}

// ---------------------------------------------------------------------------
// Fused SAGE layer GEMM: out = act( (agg/max(cnt,1)) @ Wl + xroot @ Wr + b ).
// One wave32 computes a 16x64 output strip (4 accumulators) with
// V_WMMA_F32_16X16X4_F32; A fragments (mean-path + root-path) are loaded once
// per K-step and reused across the 4 N-tiles -> 8 WMMAs per A load.
// ISA fragment layouts (CDNA5 ISA):
//   A 16x4 f32: lanes 0-15 hold K={k0,k0+1}, lanes 16-31 hold K={k0+2,k0+3}
//   B 4x16 f32: same K split; lane%16 selects the N column
//   C/D 16x16 f32: VGPR r -> row tm+r (lanes 0-15) / tm+8+r (lanes 16-31)
// ---------------------------------------------------------------------------
template <int K, int NCOL, bool RELU>
__global__ void __launch_bounds__(256)
sage_gemm(const float* __restrict__ agg, const float* __restrict__ cnt,
          const float* __restrict__ xroot, const float* __restrict__ Wl,
          const float* __restrict__ Wr, const float* __restrict__ bias,
          float* __restrict__ out, int nrows)
{
    constexpr int JT     = 4;                 // 4 x 16 = 64 columns per wave
    constexpr int GROUPS = NCOL / (16 * JT);  // N-strips per row block

    const int lane = threadIdx.x & 31;
    const int half = lane >> 4;     // 0: K pair {0,1}, 1: K pair {2,3}
    const int l    = lane & 15;

    const int wid = blockIdx.x * (blockDim.x >> 5) + (threadIdx.x >> 5);
    const int tm  = (wid / GROUPS) * 16;
    const int tn  = (wid % GROUPS) * (16 * JT);
    if (tm >= nrows) return;        // uniform per wave: EXEC stays all-ones

    const int row = tm + l;
    const float  inv  = 1.0f / fmaxf(cnt[row], 1.0f);   // mean divisor
    const float* aRow = agg   + (size_t)row * K;
    const float* xRow = xroot + (size_t)row * K;

    v8f acc[JT] = {};
#pragma unroll 2
    for (int k0 = 0; k0 < K; k0 += 4) {
        const int ka = k0 + 2 * half;

        v2f am, ar;
        am.x = aRow[ka] * inv;  am.y = aRow[ka + 1] * inv;   // mean path
        ar.x = xRow[ka];        ar.y = xRow[ka + 1];         // root path

#pragma unroll
        for (int j = 0; j < JT; ++j) {
            const int col = tn + 16 * j + l;
            v2f bl, br;
            bl.x = Wl[(size_t)ka * NCOL + col];
            bl.y = Wl[(size_t)(ka + 1) * NCOL + col];
            br.x = Wr[(size_t)ka * NCOL + col];
            br.y = Wr[(size_t)(ka + 1) * NCOL + col];

            acc[j] = __builtin_amdgcn_wmma_f32_16x16x4_f32(
                false, am, false, bl, (short)0, acc[j], false, false);
            acc[j] = __builtin_amdgcn_wmma_f32_16x16x4_f32(
                false, ar, false, br, (short)0, acc[j], false, false);
        }
    }

#pragma unroll
    for (int j = 0; j < JT; ++j) {
        const int col = tn + 16 * j + l;
        const float bv = bias[col];
#pragma unroll
        for (int r = 0; r < 8; ++r) {
            float v = acc[j][r] + bv;
            if (RELU) v = fmaxf(v, 0.0f);
            out[(size_t)(tm + 8 * half + r) * NCOL + col] = v;
        }
    }
}

// ---------------------------------------------------------------------------
// Host side
// ---------------------------------------------------------------------------
extern "C" void kernel_launch(void* const* d_in, const int* in_sizes, int n_in,
                              void* d_out, int out_size, void* d_ws, size_t ws_size,
                              hipStream_t stream)
{
    constexpr int IN_DIM = 128, HID = 256, EMB = 128;

    const float* x   = (const float*)d_in[0];
    const int*   ei  = (const int*)  d_in[1];
    const float* W1l = (const float*)d_in[2];
    const float* b1  = (const float*)d_in[3];
    const float* W1r = (const float*)d_in[4];
    const float* W2l = (const float*)d_in[5];
    const float* b2  = (const float*)d_in[6];
    const float* W2r = (const float*)d_in[7];

    const int N = in_sizes[0] / IN_DIM;
    const int E = in_sizes[1] / 2;
    const int* src = ei;          // edge_index[0]
    const int* dst = ei + E;      // edge_index[1]

    // workspace layout (floats): cnt[N] | agg1[N*128] | h[N*256] | agg2[N*256]
    float* cnt  = (float*)d_ws;
    float* agg1 = cnt  + (((size_t)N + 63) & ~(size_t)63);
    float* h    = agg1 + (size_t)N * IN_DIM;
    float* agg2 = h    + (size_t)N * HID;

    hipMemsetAsync(cnt,  0, sizeof(float) * (size_t)N,          stream);
    hipMemsetAsync(agg1, 0, sizeof(float) * (size_t)N * IN_DIM, stream);
    hipMemsetAsync(agg2, 0, sizeof(float) * (size_t)N * HID,    stream);

    const dim3 blk(256);
    const int scatterBlocks = 2048;   // 16384 waves grid-striding over edges
    const int tmN = (N + 15) / 16;

    // ---- layer 1 ----
    sage_scatter<<<scatterBlocks, blk, 0, stream>>>(x, src, dst, agg1, cnt,
                                                    E, IN_DIM, 1);
    const int waves_l1  = tmN * (HID / 64);
    const int blocks_l1 = (waves_l1 + 7) / 8;
    sage_gemm<IN_DIM, HID, true><<<blocks_l1, blk, 0, stream>>>(
        agg1, cnt, x, W1l, W1r, b1, h, N);

    // ---- layer 2 ----
    sage_scatter<<<scatterBlocks, blk, 0, stream>>>(h, src, dst, agg2, cnt,
                                                    E, HID, 0);
    const int waves_l2  = tmN * (EMB / 64);
    const int blocks_l2 = (waves_l2 + 7) / 8;
    sage_gemm<HID, EMB, false><<<blocks_l2, blk, 0, stream>>>(
        agg2, cnt, h, W2l, W2r, b2, (float*)d_out, N);
}